// BitInput_1632087573154
// MI455X (gfx1250) — compile-verified
//
#include <hip/hip_runtime.h>
#include <stdint.h>

#define BIT_SIZE      256
#define WAVE_ELEMS    8                    // (b,f) elements per wave
#define BLOCK_THREADS 256                  // 8 wave32's
#define BLOCK_ELEMS   64                   // 8 waves * 8 elements

typedef __attribute__((ext_vector_type(4))) float v4f;

#if defined(__AMDGCN__)
typedef __attribute__((address_space(1))) int gint_t;   // global
typedef __attribute__((address_space(3))) int lint_t;   // LDS
#endif

// 6-VALU integer hash (lowbias32 constants): counter-based, deterministic
// per output element index -> graph-replay safe.
__device__ __forceinline__ uint32_t hash32(uint32_t x) {
    x ^= x >> 16; x *= 0x21f0aaadu;
    x ^= x >> 15; x *= 0xd35a2d97u;
    x ^= x >> 15;
    return x;
}

// One element's 8 bits for this lane: k = 4*lane + 128*j + q.
// thr = p * 2^32, so (hash < thr) <=> (u < p) with no per-output shift.
__device__ __forceinline__ void emit8(float* __restrict__ outp,
                                      uint32_t ctr, uint32_t thr) {
    #pragma unroll
    for (int j = 0; j < 2; ++j) {
        v4f v;
        #pragma unroll
        for (int q = 0; q < 4; ++q) {
            uint32_t h = hash32(ctr + (uint32_t)(128 * j + q));
            v[q] = (h < thr) ? 1.0f : 0.0f;
        }
        __builtin_nontemporal_store(v, (v4f*)(outp + 128 * j));
    }
}

__global__ __launch_bounds__(BLOCK_THREADS) void bernoulli_bits_kernel(
    const float* __restrict__ p, float* __restrict__ out, int n_elems) {
    __shared__ __align__(16) float sp[BLOCK_ELEMS];

    const int tid        = threadIdx.x;
    const int lane       = tid & 31;      // wave32
    const int wave       = tid >> 5;
    const int blockElem0 = blockIdx.x * BLOCK_ELEMS;

    // ---- Stage this block's 64 probabilities into LDS via the CDNA5
    // ---- async global->LDS path (ASYNCcnt), portable fallback otherwise.
#if defined(__gfx1250__) && __has_builtin(__builtin_amdgcn_global_load_async_to_lds_b32)
    if (tid < BLOCK_ELEMS && (blockElem0 + tid) < n_elems) {
        const float* gp = p + blockElem0 + tid;
        __builtin_amdgcn_global_load_async_to_lds_b32(
            (gint_t*)(uintptr_t)gp, (lint_t*)&sp[tid],
            /*imm offset*/0, /*cpol*/0);
    }
  #if __has_builtin(__builtin_amdgcn_s_wait_asynccnt)
    __builtin_amdgcn_s_wait_asynccnt(0);
  #else
    asm volatile("s_wait_asynccnt 0" ::: "memory");
  #endif
#else
    if (tid < BLOCK_ELEMS && (blockElem0 + tid) < n_elems)
        sp[tid] = p[blockElem0 + tid];
#endif
    __syncthreads();   // gfx1250: s_barrier_signal / s_barrier_wait

    const int ebase = blockElem0 + wave * WAVE_ELEMS;
    if (ebase >= n_elems) return;

    // One base pointer; all 16 b128 NT stores use immediate offsets from it.
    float*         outp  = out + (size_t)ebase * BIT_SIZE + 4 * lane;
    const uint32_t cbase = (uint32_t)ebase * BIT_SIZE + 4u * (uint32_t)lane;

    if (ebase + WAVE_ELEMS <= n_elems) {
        // Fast path: two ds_load_b128 pull this wave's 8 probabilities.
        const v4f p0 = *(const v4f*)&sp[wave * WAVE_ELEMS];
        const v4f p1 = *(const v4f*)&sp[wave * WAVE_ELEMS + 4];
        #pragma unroll
        for (int e = 0; e < WAVE_ELEMS; ++e) {
            const float    pv  = (e < 4) ? p0[e & 3] : p1[e & 3];
            const uint32_t thr = (uint32_t)(pv * 4294967296.0f);
            emit8(outp + e * BIT_SIZE, cbase + (uint32_t)(e * BIT_SIZE), thr);
        }
    } else {
        for (int e = 0; e < WAVE_ELEMS && (ebase + e) < n_elems; ++e) {
            const uint32_t thr =
                (uint32_t)(sp[wave * WAVE_ELEMS + e] * 4294967296.0f);
            emit8(outp + e * BIT_SIZE, cbase + (uint32_t)(e * BIT_SIZE), thr);
        }
    }
}

extern "C" void kernel_launch(void* const* d_in, const int* in_sizes, int n_in,
                              void* d_out, int out_size, void* d_ws, size_t ws_size,
                              hipStream_t stream) {
    const float* p   = (const float*)d_in[0];
    float*       out = (float*)d_out;
    const int n      = in_sizes[0];                      // 1024*512 = 524288
    const int blocks = (n + BLOCK_ELEMS - 1) / BLOCK_ELEMS;
    bernoulli_bits_kernel<<<blocks, BLOCK_THREADS, 0, stream>>>(p, out, n);
}